// UnimodalConcentratedLossWithWarmup_64939905516213
// MI455X (gfx1250) — compile-verified
//
#include <hip/hip_runtime.h>
#include <cstdint>

typedef unsigned int       u32;
typedef unsigned long long u64;
typedef u32 v4u __attribute__((ext_vector_type(4)));
typedef int  v4i __attribute__((ext_vector_type(4)));
typedef int  v8i __attribute__((ext_vector_type(8)));

#define BATCH      262144
#define NCLS       101
#define BLOCKS     1024
#define WAVES      8
#define THREADS    (WAVES * 32)
#define ROWS_PER_BLOCK (BATCH / BLOCKS)        /* 256 */
#define TILE_ROWS  64
#define TILES_PER_BLOCK (ROWS_PER_BLOCK / TILE_ROWS) /* 4 */

// ---------------- wave32 reductions ----------------
__device__ __forceinline__ float wave_sum(float v) {
    v += __shfl_xor(v, 1, 32);
    v += __shfl_xor(v, 2, 32);
    v += __shfl_xor(v, 4, 32);
    v += __shfl_xor(v, 8, 32);
    v += __shfl_xor(v, 16, 32);
    return v;
}
__device__ __forceinline__ float wave_max(float v) {
    v = fmaxf(v, __shfl_xor(v, 1, 32));
    v = fmaxf(v, __shfl_xor(v, 2, 32));
    v = fmaxf(v, __shfl_xor(v, 4, 32));
    v = fmaxf(v, __shfl_xor(v, 8, 32));
    v = fmaxf(v, __shfl_xor(v, 16, 32));
    return v;
}

// Low 32 bits of a generic pointer to __shared__ == byte offset in LDS
// (addrspacecast local->flat only ORs the shared aperture into the high half).
__device__ __forceinline__ u32 lds_byte_offset(const void* p) {
    return (u32)(u64)p;
}

// Issue a TDM 2-D tile load: TILE_ROWS rows x NCLS f32 from global -> LDS.
// Descriptor per CDNA5 ISA (group0 128b, group1 256b).
// This toolchain's builtin takes 6 args (extra int32x8 descriptor group).
__device__ __forceinline__ void tdm_load_tile(const float* gsrc, float* ldst,
                                              int tile_rows) {
    u64 ga = (u64)gsrc;
    u32 la = lds_byte_offset(ldst);

    v4u g0;
    g0.x = 1u;                                    // count=1 (valid), user mode
    g0.y = la;                                    // lds_addr [63:32]
    g0.z = (u32)ga;                               // global_addr [95:64]
    g0.w = (u32)((ga >> 32) & 0x01FFFFFFu)        // global_addr [120:96]
         | (2u << 30);                            // type=2 ("image")

    v8i g1;
    g1[0] = (int)(2u << 16);                      // data_size=2 -> 4 bytes
    g1[1] = (int)((u32)NCLS << 16);               // tensor_dim0[15:0] @ bits 63:48
    g1[2] = (int)(((u32)(BATCH & 0xFFFF)) << 16); // dim0 hi=0; tensor_dim1[15:0]
    g1[3] = (int)(((u32)(BATCH >> 16) & 0xFFFF)   // tensor_dim1[31:16]
         | ((u32)NCLS << 16));                    // tile_dim0 @ bits 127:112
    g1[4] = (int)((u32)tile_rows & 0xFFFF);       // tile_dim1; tile_dim2=0
    g1[5] = (int)NCLS;                            // tensor_dim0_stride[31:0]
    g1[6] = 0;                                    // stride0 hi; stride1 lo
    g1[7] = 0;                                    // stride1 hi (2-D: unused)

    v4i z4 = {0, 0, 0, 0};
    v8i z8 = {0, 0, 0, 0, 0, 0, 0, 0};
    __builtin_amdgcn_tensor_load_to_lds(g0, g1, z4, z4, z8, 0);
}

__global__ __launch_bounds__(THREADS)
void ucl_main(const float* __restrict__ outs,
              const long long* __restrict__ tgt,
              float* __restrict__ partials) {
    __shared__ float tile[2][TILE_ROWS * NCLS];   // 51,712 B double buffer
    __shared__ float sred[3][WAVES];

    const int wave = threadIdx.x >> 5;
    const int lane = threadIdx.x & 31;
    const long long row0 = (long long)blockIdx.x * ROWS_PER_BLOCK;

    float acc_ce = 0.0f, acc_conc = 0.0f, acc_uni = 0.0f;

    if (wave == 0)
        tdm_load_tile(outs + row0 * NCLS, tile[0], TILE_ROWS);

    for (int t = 0; t < TILES_PER_BLOCK; ++t) {
        if (wave == 0) {
            if (t + 1 < TILES_PER_BLOCK) {
                tdm_load_tile(outs + (row0 + (long long)(t + 1) * TILE_ROWS) * NCLS,
                              tile[(t + 1) & 1], TILE_ROWS);
                __builtin_amdgcn_s_wait_tensorcnt(1); // oldest (tile t) done
            } else {
                __builtin_amdgcn_s_wait_tensorcnt(0);
            }
        }
        __syncthreads();                          // tile t visible to all waves

        const float* buf = tile[t & 1];
        for (int rl = wave; rl < TILE_ROWS; rl += WAVES) {
            const long long r = row0 + (long long)t * TILE_ROWS + rl;
            const float* rowp = buf + rl * NCLS;
            const int j0 = lane * 4;

            float x[4];
            #pragma unroll
            for (int k = 0; k < 4; ++k) {
                int j = j0 + k;
                x[k] = (j < NCLS) ? rowp[j] : -3.402823466e38f;
            }
            float m = wave_max(fmaxf(fmaxf(x[0], x[1]), fmaxf(x[2], x[3])));

            float e[4], se = 0.0f, sej = 0.0f;
            #pragma unroll
            for (int k = 0; k < 4; ++k) {
                e[k] = __expf(x[k] - m);
                se  += e[k];
                sej += e[k] * (float)(j0 + k);
            }
            se  = wave_sum(se);
            sej = wave_sum(sej);
            float inv  = 1.0f / se;
            float pred = sej * inv;

            float vnum = 0.0f;
            #pragma unroll
            for (int k = 0; k < 4; ++k) {
                float c = (float)(j0 + k) - pred;
                vnum += e[k] * c * c;
            }
            vnum = wave_sum(vnum);
            float var = fmaxf(vnum * inv, 1e-6f);
            float lse = __logf(se);

            const long long tv = tgt[r];
            const int   ti = (int)tv;
            const float tf = (float)ti;

            #pragma unroll
            for (int k = 0; k < 4; ++k)
                if (j0 + k == ti) acc_ce += (x[k] - m - lse); // logp[target]

            if (lane == 0) {
                float err = pred - tf; err *= err;
                acc_conc += 0.5f * __logf(var) + err / (2.0f * var);
            }

            float p[4];
            #pragma unroll
            for (int k = 0; k < 4; ++k) p[k] = e[k] * inv;
            float pnext = __shfl_down(p[0], 1, 32);
            #pragma unroll
            for (int k = 0; k < 4; ++k) {
                int j = j0 + k;
                if (j < NCLS - 1) {               // j <= 99
                    float d = p[k] - ((k < 3) ? p[k + 1] : pnext);
                    // sign=-1 if j<target else +1; pen = relu(-d*sign)
                    acc_uni += (j < ti) ? fmaxf(0.0f, d) : fmaxf(0.0f, -d);
                }
            }
        }
        __syncthreads();                          // all reads of buf done
    }

    acc_ce  = wave_sum(acc_ce);
    acc_uni = wave_sum(acc_uni);
    if (lane == 0) {
        sred[0][wave] = acc_ce;
        sred[1][wave] = acc_conc;
        sred[2][wave] = acc_uni;
    }
    __syncthreads();
    if (threadIdx.x == 0) {
        float a0 = 0.0f, a1 = 0.0f, a2 = 0.0f;
        #pragma unroll
        for (int w = 0; w < WAVES; ++w) {
            a0 += sred[0][w]; a1 += sred[1][w]; a2 += sred[2][w];
        }
        partials[3 * blockIdx.x + 0] = a0;
        partials[3 * blockIdx.x + 1] = a1;
        partials[3 * blockIdx.x + 2] = a2;
    }
}

__global__ __launch_bounds__(256)
void ucl_finalize(const float* __restrict__ partials, float* __restrict__ out) {
    __shared__ float s[3][256];
    const int tid = threadIdx.x;
    float a0 = 0.0f, a1 = 0.0f, a2 = 0.0f;
    for (int b = tid; b < BLOCKS; b += 256) {
        a0 += partials[3 * b + 0];
        a1 += partials[3 * b + 1];
        a2 += partials[3 * b + 2];
    }
    s[0][tid] = a0; s[1][tid] = a1; s[2][tid] = a2;
    __syncthreads();
    for (int off = 128; off >= 1; off >>= 1) {
        if (tid < off) {
            s[0][tid] += s[0][tid + off];
            s[1][tid] += s[1][tid + off];
            s[2][tid] += s[2][tid + off];
        }
        __syncthreads();
    }
    if (tid == 0) {
        const float invB = 1.0f / (float)BATCH;
        float ce   = -s[0][0] * invB;             // ce_weight = 1.0 (epoch 0)
        float conc =  s[1][0] * invB;
        float uni  = 10.0f * (s[2][0] * invB);    // lambda = 10.0 (epoch 0)
        out[0] = conc + uni + ce;                 // total
        out[1] = conc;                            // concentrated_loss
        out[2] = uni;                             // weighted_unimodal_loss
    }
}

extern "C" void kernel_launch(void* const* d_in, const int* in_sizes, int n_in,
                              void* d_out, int out_size, void* d_ws, size_t ws_size,
                              hipStream_t stream) {
    (void)in_sizes; (void)n_in; (void)out_size; (void)ws_size;
    const float*     outs = (const float*)d_in[0];
    const long long* tgt  = (const long long*)d_in[1];
    float* partials = (float*)d_ws;               // BLOCKS*3 floats = 12 KB
    float* out      = (float*)d_out;              // 3 floats

    ucl_main<<<BLOCKS, THREADS, 0, stream>>>(outs, tgt, partials);
    ucl_finalize<<<1, 256, 0, stream>>>(partials, out);
}